// RecurrentNALU_53128745452243
// MI455X (gfx1250) — compile-verified
//
#include <hip/hip_runtime.h>

// RecurrentNALU single step, fused: two f32 WMMA GEMMs (gate + NAC add path)
// + VALU MNAC product path, all staged through LDS. gfx1250 / wave32.

typedef float v2f __attribute__((ext_vector_type(2)));
typedef float v8f __attribute__((ext_vector_type(8)));

#define B_TOTAL 4096
#define IN_SZ   128
#define HID     128
#define CAT     256          // IN_SZ + HID
#define TILE_B  16
#define WM_PITCH 260         // 256 + 4 floats pad -> W reads rotate banks by 4

__global__ __launch_bounds__(256)
void nalu_step_kernel(const float* __restrict__ x_t,
                      const float* __restrict__ h_tm1,
                      const float* __restrict__ W_add,
                      const float* __restrict__ W_mul,
                      const float* __restrict__ G_add,
                      float* __restrict__ out)
{
    extern __shared__ float smem[];
    float* x_lds  = smem;                 // [TILE_B][CAT]        16 KB
    float* wm_lds = smem + TILE_B * CAT;  // [HID][WM_PITCH]     130 KB

    const int tid   = threadIdx.x;
    const int bbase = blockIdx.x * TILE_B;

    // ---- Stage x tile (concat of x_t | h_tm1), coalesced ----
    for (int i = tid; i < TILE_B * CAT; i += 256) {
        const int row = i >> 8;           // /CAT
        const int col = i & (CAT - 1);
        float v = (col < IN_SZ) ? x_t[(bbase + row) * IN_SZ + col]
                                : h_tm1[(bbase + row) * HID + (col - IN_SZ)];
        x_lds[i] = v;
    }
    // ---- Stage W_mul clamped to [0,1], row-padded ----
    for (int i = tid; i < HID * CAT; i += 256) {
        const int h = i >> 8;
        const int k = i & (CAT - 1);
        float w = W_mul[i];
        w = fminf(fmaxf(w, 0.0f), 1.0f);
        wm_lds[h * WM_PITCH + k] = w;
    }
    __syncthreads();

    const int lane  = tid & 31;
    const int wave  = tid >> 5;           // 0..7 -> h tile
    const int h0    = wave * 16;
    const int n     = lane & 15;          // N index (column within h tile)
    const int h     = h0 + n;
    const int half  = lane >> 4;          // 0: lanes 0-15, 1: lanes 16-31
    const int khalf = half << 1;          // A/B K offset per 32-bit WMMA layout
    const int m     = lane & 15;          // A-matrix row (M = lane%16)

    // ---- GEMM phase: g_lin = x@G^T, a = x@clip(W_add)^T via V_WMMA_F32_16X16X4_F32 ----
    v8f acc_g = {0.f,0.f,0.f,0.f,0.f,0.f,0.f,0.f};
    v8f acc_a = {0.f,0.f,0.f,0.f,0.f,0.f,0.f,0.f};

    const float* Arow = &x_lds[m * CAT];
    const float* Grow = &G_add[h * CAT];
    const float* Wrow = &W_add[h * CAT];

#pragma unroll 8
    for (int k0 = 0; k0 < CAT; k0 += 4) {
        // A 16x4 f32: lanes 0-15 hold K={k0,k0+1}, lanes 16-31 K={k0+2,k0+3}
        v2f afrag = *(const v2f*)(Arow + k0 + khalf);
        // B 4x16 f32 (mirrored layout): B[k,n] = Weight[h0+n, k]
        v2f bg = *(const v2f*)(Grow + k0 + khalf);
        v2f bw = *(const v2f*)(Wrow + k0 + khalf);
        bw.x = fminf(fmaxf(bw.x, -1.0f), 1.0f);
        bw.y = fminf(fmaxf(bw.y, -1.0f), 1.0f);
        acc_g = __builtin_amdgcn_wmma_f32_16x16x4_f32(false, afrag, false, bg,
                                                      (short)0, acc_g, false, false);
        acc_a = __builtin_amdgcn_wmma_f32_16x16x4_f32(false, afrag, false, bw,
                                                      (short)0, acc_a, false, false);
    }

    // ---- MNAC phase: p[r] = prod_k (x[b_r,k]*W[k,h] + 1 - W[k,h]) ----
    // C/D layout: VGPR r holds M = r + 8*half, N = lane%16  -> b_r = bbase + rbase + r
    const int rbase = half * 8;
    float p[8] = {1.f,1.f,1.f,1.f,1.f,1.f,1.f,1.f};
    const float* wrow = &wm_lds[h * WM_PITCH];

    for (int kc = 0; kc < CAT; kc += 4) {
        const float4 w   = *(const float4*)(wrow + kc);                 // b128, bank-rotated
        const float4 omw = make_float4(1.f - w.x, 1.f - w.y, 1.f - w.z, 1.f - w.w);
#pragma unroll
        for (int r = 0; r < 8; ++r) {
            const float4 xv = *(const float4*)(&x_lds[(rbase + r) * CAT + kc]); // broadcast
            float t0 = fmaf(xv.x, w.x, omw.x);
            float t1 = fmaf(xv.y, w.y, omw.y);
            float t2 = fmaf(xv.z, w.z, omw.z);
            float t3 = fmaf(xv.w, w.w, omw.w);
            p[r] *= (t0 * t1) * (t2 * t3);
        }
    }

    // ---- Blend: y = sigmoid(g_lin)*a + (1-sigmoid)*m ----
#pragma unroll
    for (int r = 0; r < 8; ++r) {
        const float gl = acc_g[r];
        const float e  = __builtin_amdgcn_exp2f(gl * -1.44269504f);     // exp(-gl)
        const float g  = __builtin_amdgcn_rcpf(1.0f + e);               // sigmoid
        const float y  = g * acc_a[r] + (1.0f - g) * p[r];
        out[(bbase + rbase + r) * HID + h] = y;
    }
}

extern "C" void kernel_launch(void* const* d_in, const int* in_sizes, int n_in,
                              void* d_out, int out_size, void* d_ws, size_t ws_size,
                              hipStream_t stream) {
    const float* x_t   = (const float*)d_in[0];
    const float* h_tm1 = (const float*)d_in[1];
    const float* W_add = (const float*)d_in[2];
    const float* W_mul = (const float*)d_in[3];
    const float* G_add = (const float*)d_in[4];
    float* out = (float*)d_out;

    const size_t lds_bytes = (size_t)(TILE_B * CAT + HID * WM_PITCH) * sizeof(float); // ~146 KB
    nalu_step_kernel<<<B_TOTAL / TILE_B, 256, lds_bytes, stream>>>(
        x_t, h_tm1, W_add, W_mul, G_add, out);
}